// FunctionSet_47467978555720
// MI455X (gfx1250) — compile-verified
//
#include <hip/hip_runtime.h>
#include <stdint.h>

typedef float v2f __attribute__((ext_vector_type(2)));
typedef float v8f __attribute__((ext_vector_type(8)));

#define ROTL32(x, r) (((x) << (r)) | ((x) >> (32 - (r))))

// JAX threefry2x32 with key = jax.random.key(42) -> (k0,k1) = (0, 42)
__device__ __forceinline__ void threefry2x32_42(uint32_t c0, uint32_t c1,
                                                uint32_t& o0, uint32_t& o1) {
  const uint32_t ks0 = 0u;
  const uint32_t ks1 = 42u;
  const uint32_t ks2 = 0x1BD11BDAu ^ ks0 ^ ks1;  // 0x1BD11BF0
  uint32_t x0 = c0 + ks0;
  uint32_t x1 = c1 + ks1;
#define TFR(r) { x0 += x1; x1 = ROTL32(x1, r); x1 ^= x0; }
  TFR(13) TFR(15) TFR(26) TFR(6)   x0 += ks1; x1 += ks2 + 1u;
  TFR(17) TFR(29) TFR(16) TFR(24)  x0 += ks2; x1 += ks0 + 2u;
  TFR(13) TFR(15) TFR(26) TFR(6)   x0 += ks0; x1 += ks1 + 3u;
  TFR(17) TFR(29) TFR(16) TFR(24)  x0 += ks1; x1 += ks2 + 4u;
  TFR(13) TFR(15) TFR(26) TFR(6)   x0 += ks2; x1 += ks0 + 5u;
#undef TFR
  o0 = x0; o1 = x1;
}

__device__ __forceinline__ float lane_bcast(float v, int srcLane) {
  return __int_as_float(
      __builtin_amdgcn_ds_bpermute(srcLane << 2, __float_as_int(v)));
}

// Fetch C[M = m][N = srcLane-col] where value lives in lane srcLane, VGPR idx.
__device__ __forceinline__ float gather_c(const v8f& C, int srcLane, int idx) {
  float r = lane_bcast(C[0], srcLane);
#pragma unroll
  for (int k = 1; k < 8; ++k) {
    float g = lane_bcast(C[k], srcLane);
    r = (idx == k) ? g : r;
  }
  return r;
}

__global__ void __launch_bounds__(256) ifs_wmma_kernel(
    const float* __restrict__ points, const float* __restrict__ probs,
    const float* __restrict__ A, const float* __restrict__ b,
    const float* __restrict__ var_w, const float* __restrict__ colors,
    float* __restrict__ out, uint32_t half_n, uint32_t h) {
  __shared__ float s_logit[16], s_w0[16], s_w1[16], s_col[16];
  const int tid = threadIdx.x;
  if (tid < 16) {
    s_logit[tid] = logf(probs[tid]);
    s_w0[tid] = var_w[2 * tid];
    s_w1[tid] = var_w[2 * tid + 1];
    s_col[tid] = colors[tid];
  }
  __syncthreads();

  const int lane = tid & 31;
  const int fn = lane & 15;
  const bool lo = (lane < 16);

  // B operands: all 16 functions packed as columns.
  // Rows: k0 = a_i0 (times x), k1 = a_i1 (times y), k2 = b_i (times 1), k3 = 0.
  // f32 4x16 B layout: VGPR0 = rows {k0 | k1}, VGPR1 = rows {k2 | k3}.
  v2f Btx, Bty;
  Btx.x = lo ? A[4 * fn + 0] : A[4 * fn + 1];
  Btx.y = lo ? b[2 * fn + 0] : 0.0f;
  Bty.x = lo ? A[4 * fn + 2] : A[4 * fn + 3];
  Bty.y = lo ? b[2 * fn + 1] : 0.0f;

  const uint32_t p1 = blockIdx.x * 256u + (uint32_t)tid;  // first-half point
  const uint32_t p2 = p1 + half_n;                        // second-half point

  // ---- RNG: gumbel-argmax choices for p1 (word0) and p2 (word1) ----
  const uint32_t base = p1 * 16u;
  float best1 = -INFINITY, best2 = -INFINITY;
  int c1 = 0, c2 = 0;
#pragma unroll
  for (int j = 0; j < 16; ++j) {
    uint32_t w0, w1;
    threefry2x32_42(base + (uint32_t)j, base + (uint32_t)j + h, w0, w1);
    float u0 = __uint_as_float((w0 >> 9) | 0x3f800000u) - 1.0f;
    float u1 = __uint_as_float((w1 >> 9) | 0x3f800000u) - 1.0f;
    u0 = fmaxf(u0, 1.17549435e-38f);
    u1 = fmaxf(u1, 1.17549435e-38f);
    float v0 = s_logit[j] - __logf(-__logf(u0));
    float v1 = s_logit[j] - __logf(-__logf(u1));
    if (v0 > best1) { best1 = v0; c1 = j; }
    if (v1 > best2) { best2 = v1; c2 = j; }
  }

  const int m = lane & 15;
  const int idx = lane & 7;
  const v8f zero = {};

#pragma unroll
  for (int g = 0; g < 2; ++g) {
    const uint32_t p = (g == 0) ? p1 : p2;
    const int c = (g == 0) ? c1 : c2;

    const float x = points[3 * p + 0];
    const float y = points[3 * p + 1];
    const float pc = points[3 * p + 2];

    // A operands. Tile0 = wave points 0..15, tile1 = wave points 16..31.
    // f32 16x4 A layout: lanes 0-15 hold K=0,1 (x,y); lanes 16-31 hold K=2,3 (1,0).
    const float xo = lane_bcast(x, lane + 16);  // partner lane's x (for tile1)
    const float yo = lane_bcast(y, lane + 16);
    v2f a0, a1;
    a0.x = lo ? x : 1.0f;  a0.y = lo ? y : 0.0f;
    a1.x = lo ? xo : 1.0f; a1.y = lo ? yo : 0.0f;

    // C[m][f] = t-component of point m under function f, for all 16 functions.
    v8f Ctx0 = __builtin_amdgcn_wmma_f32_16x16x4_f32(false, a0, false, Btx,
                                                     (short)0, zero, false, false);
    v8f Cty0 = __builtin_amdgcn_wmma_f32_16x16x4_f32(false, a0, false, Bty,
                                                     (short)0, zero, false, false);
    v8f Ctx1 = __builtin_amdgcn_wmma_f32_16x16x4_f32(false, a1, false, Btx,
                                                     (short)0, zero, false, false);
    v8f Cty1 = __builtin_amdgcn_wmma_f32_16x16x4_f32(false, a1, false, Bty,
                                                     (short)0, zero, false, false);

    // Gather this lane's point: element (M=m, N=c) -> lane c+16*(m>=8), VGPR m&7.
    const int srcLane = c + ((m & 8) ? 16 : 0);
    const float tx0 = gather_c(Ctx0, srcLane, idx);
    const float tx1 = gather_c(Ctx1, srcLane, idx);
    const float ty0 = gather_c(Cty0, srcLane, idx);
    const float ty1 = gather_c(Cty1, srcLane, idx);
    const float tx = lo ? tx0 : tx1;
    const float ty = lo ? ty0 : ty1;

    const float w0 = s_w0[c], w1 = s_w1[c];
    out[3 * p + 0] = w0 * tx + w1 * __sinf(tx);
    out[3 * p + 1] = w0 * ty + w1 * __sinf(ty);
    out[3 * p + 2] = (pc + s_col[c]) * 0.5f;
  }
}

extern "C" void kernel_launch(void* const* d_in, const int* in_sizes, int n_in,
                              void* d_out, int out_size, void* d_ws, size_t ws_size,
                              hipStream_t stream) {
  const float* points = (const float*)d_in[0];
  const float* probs  = (const float*)d_in[1];
  const float* A      = (const float*)d_in[2];
  const float* b      = (const float*)d_in[3];
  const float* var_w  = (const float*)d_in[4];
  const float* colors = (const float*)d_in[5];
  float* out = (float*)d_out;

  const uint32_t n = (uint32_t)(in_sizes[0] / 3);        // 8388608
  const uint32_t half_n = n / 2u;                        // 4194304 pairs
  const uint32_t h = n * 8u;                             // 16*n/2: counter offset
  const uint32_t blocks = (half_n + 255u) / 256u;        // exact: 16384

  ifs_wmma_kernel<<<blocks, 256, 0, stream>>>(points, probs, A, b, var_w,
                                              colors, out, half_n, h);
}